// FuzzyNNWithReducedRules_9474697854979
// MI455X (gfx1250) — compile-verified
//
#include <hip/hip_runtime.h>
#include <math.h>
#include <stdint.h>

typedef float v2f __attribute__((ext_vector_type(2)));
typedef float v8f __attribute__((ext_vector_type(8)));

#define NROWS  131072
#define F      32
#define M      64
#define LATENT 128
#define OUTN   128
#define TILE   128      // rows per block
#define OS     68       // padded stride of o_r tile (bank-conflict-free A frags)
#define W1S    66       // padded stride of W1^T rows
#define HS     132      // padded stride of h tile rows
#define EPS    1e-5f

__global__ __launch_bounds__(256) void fuzzy_fused_kernel(
    const float* __restrict__ x,     const float* __restrict__ means,
    const float* __restrict__ sigmas,const float* __restrict__ W1,
    const float* __restrict__ b1,    const float* __restrict__ W2,
    const float* __restrict__ b2,    float* __restrict__ out)
{
    __shared__ float s_x[TILE * F];           // 16 KB : x tile
    __shared__ float s_o[TILE * OS];          // 34 KB : o_r then normalized A
    __shared__ float s_w1[LATENT * W1S];      // 33 KB : B1[k][n] = W1[n][k], stored [n][k] padded
    __shared__ float s_h[8 * 16 * HS];        // 66 KB : per-wave tanh activations (A for GEMM2)
    __shared__ float s_b1[LATENT];
    __shared__ float s_b2[OUTN];

    const int t = threadIdx.x;

    // ---- kick off async global->LDS fill of the x tile (ASYNCcnt, no VGPR staging) ----
    // GLOBAL_LOAD_ASYNC_TO_LDS_B128, GVS mode: LDS address taken from the vdst VGPR, mem = SGPR base + VGPR off
    {
        const float* gx = x + (size_t)blockIdx.x * TILE * F;
        unsigned lds_base = (unsigned)(uintptr_t)(&s_x[0]);
#pragma unroll
        for (int k = 0; k < 4; ++k) {                 // 256 threads x 4 x 16B = 16 KB
            unsigned off = (unsigned)(t * 16 + k * 4096);
            unsigned dsa = lds_base + off;
            asm volatile("global_load_async_to_lds_b128 %0, %1, %2"
                         :: "v"(dsa), "v"(off), "s"(gx)
                         : "memory");
        }
    }

    // ---- per-thread membership params into registers (overlaps with async copy) ----
    const int m = t & 63;
    float rm[F], ri[F];
#pragma unroll
    for (int f = 0; f < F; ++f) {
        rm[f] = means[f * M + m];
        float sg = sigmas[f * M + m];
        ri[f] = 1.0f / (sg * sg);
    }

    // ---- cooperative staging of W1 (padded) and biases ----
    for (int idx = t; idx < LATENT * M; idx += 256) {
        int n = idx >> 6, k = idx & 63;
        s_w1[n * W1S + k] = W1[idx];          // W1 is (LATENT, M) row-major
    }
    if (t < LATENT) s_b1[t] = b1[t];
    if (t < OUTN)   s_b2[t] = b2[t];

    asm volatile("s_wait_asynccnt 0x0" ::: "memory");  // x tile landed in LDS
    __syncthreads();

    // ---- phase A: membership + product over F as exp of negative weighted SSQ ----
    const int rsub = t >> 6;
    for (int i = 0; i < 32; ++i) {
        const int row = i * 4 + rsub;
        float s = 0.0f;
#pragma unroll
        for (int f = 0; f < F; ++f) {
            float d = s_x[row * F + f] - rm[f];
            s = fmaf(d * d, ri[f], s);
        }
        s_o[row * OS + m] = __expf(-s);
    }
    __syncthreads();

    // ---- phase B: LayerNorm over M=64 (2 threads per row, pair-combine via shfl_xor) ----
    {
        const int row = t >> 1, half = t & 1;
        float* p = s_o + row * OS + half * 32;
        float sum = 0.0f, sq = 0.0f;
#pragma unroll
        for (int k = 0; k < 8; ++k) {
            float4 v = reinterpret_cast<float4*>(p)[k];
            sum += v.x + v.y + v.z + v.w;
            sq  += v.x * v.x + v.y * v.y + v.z * v.z + v.w * v.w;
        }
        sum += __shfl_xor(sum, 1, 32);
        sq  += __shfl_xor(sq, 1, 32);
        float mu = sum * (1.0f / 64.0f);
        float rs = rsqrtf(sq * (1.0f / 64.0f) - mu * mu + EPS);
#pragma unroll
        for (int k = 0; k < 8; ++k) {
            float4 v = reinterpret_cast<float4*>(p)[k];
            v.x = (v.x - mu) * rs; v.y = (v.y - mu) * rs;
            v.z = (v.z - mu) * rs; v.w = (v.w - mu) * rs;
            reinterpret_cast<float4*>(p)[k] = v;
        }
    }
    __syncthreads();

    // ---- phase C: per-wave 16-row WMMA tiles ----
    const int wave = t >> 5;
    const int lam  = t & 31;
    const int hi   = lam >> 4;       // lane half
    const int ln   = lam & 15;       // lane-in-half: A row / B,C column

    // GEMM1: (16 x 64) @ (64 x 128) via v_wmma_f32_16x16x4_f32
    v8f c[8];
#pragma unroll
    for (int nt = 0; nt < 8; ++nt) c[nt] = (v8f)(0.0f);

    const float* arow = s_o + (wave * 16 + ln) * OS;
    for (int j = 0; j < 16; ++j) {
        const int k0 = 4 * j + 2 * hi;                 // A/B frag: reg p -> k = 4j + p + 2*half
        v2f a = *(const v2f*)(arow + k0);              // ds_load_b64
#pragma unroll
        for (int nt = 0; nt < 8; ++nt) {
            const int n = nt * 16 + ln;
            v2f b = *(const v2f*)(s_w1 + n * W1S + k0);
            c[nt] = __builtin_amdgcn_wmma_f32_16x16x4_f32(
                false, a, false, b, (short)0, c[nt], false, false);
        }
    }

    // + b1, LayerNorm over LATENT=128 (rows live in 16-lane halves: xor 1/2/4/8)
    float ps[8], pq[8];
#pragma unroll
    for (int v = 0; v < 8; ++v) { ps[v] = 0.0f; pq[v] = 0.0f; }
#pragma unroll
    for (int nt = 0; nt < 8; ++nt) {
        float bias = s_b1[nt * 16 + ln];
#pragma unroll
        for (int v = 0; v < 8; ++v) {
            float e = c[nt][v] + bias;
            c[nt][v] = e;
            ps[v] += e;
            pq[v] += e * e;
        }
    }
#pragma unroll
    for (int v = 0; v < 8; ++v) {
#pragma unroll
        for (int mask = 1; mask < 16; mask <<= 1) {
            ps[v] += __shfl_xor(ps[v], mask, 32);
            pq[v] += __shfl_xor(pq[v], mask, 32);
        }
    }
    float muv[8], rsv[8];
#pragma unroll
    for (int v = 0; v < 8; ++v) {
        muv[v] = ps[v] * (1.0f / 128.0f);
        rsv[v] = rsqrtf(pq[v] * (1.0f / 128.0f) - muv[v] * muv[v] + EPS);
    }

    // tanh + stage to LDS in A-matrix order for GEMM2 (C-layout -> row-major transpose)
    float* hbase = s_h + wave * 16 * HS;
#pragma unroll
    for (int nt = 0; nt < 8; ++nt) {
#pragma unroll
        for (int v = 0; v < 8; ++v) {
            float z  = (c[nt][v] - muv[v]) * rsv[v];
            float az = fabsf(z);
            float u  = __expf(2.0f * az);
            float th = 1.0f - 2.0f / (u + 1.0f);      // tanh(|z|); stable, ->1 on overflow
            th = (z < 0.0f) ? -th : th;
            hbase[(v + 8 * hi) * HS + nt * 16 + ln] = th;
        }
    }
    // no __syncthreads needed: each wave reads back only its own s_h region

    // GEMM2: (16 x 128) @ (128 x 128); B2[k][n] = W2[n][k] contiguous float2 from L2
    v8f c2[8];
#pragma unroll
    for (int nt = 0; nt < 8; ++nt) c2[nt] = (v8f)(0.0f);

    const float* hrow = hbase + ln * HS;
    for (int j = 0; j < 32; ++j) {
        const int k0 = 4 * j + 2 * hi;
        v2f a = *(const v2f*)(hrow + k0);              // ds_load_b64, padded stride
#pragma unroll
        for (int nt = 0; nt < 8; ++nt) {
            const int n = nt * 16 + ln;
            v2f b = *(const v2f*)(W2 + n * LATENT + k0);  // global_load_b64, L2-hot
            c2[nt] = __builtin_amdgcn_wmma_f32_16x16x4_f32(
                false, a, false, b, (short)0, c2[nt], false, false);
        }
    }

    // + b2, store (coalesced 64B per 16-lane half)
    const int grow0 = blockIdx.x * TILE + wave * 16 + 8 * hi;
#pragma unroll
    for (int nt = 0; nt < 8; ++nt) {
        const int n = nt * 16 + ln;
        const float bias = s_b2[n];
#pragma unroll
        for (int v = 0; v < 8; ++v) {
            out[(size_t)(grow0 + v) * OUTN + n] = c2[nt][v] + bias;
        }
    }
}

extern "C" void kernel_launch(void* const* d_in, const int* in_sizes, int n_in,
                              void* d_out, int out_size, void* d_ws, size_t ws_size,
                              hipStream_t stream) {
    (void)in_sizes; (void)n_in; (void)out_size; (void)d_ws; (void)ws_size;
    const float* x      = (const float*)d_in[0];
    const float* means  = (const float*)d_in[1];
    const float* sigmas = (const float*)d_in[2];
    const float* W1     = (const float*)d_in[3];
    const float* b1     = (const float*)d_in[4];
    const float* W2     = (const float*)d_in[5];
    const float* b2     = (const float*)d_in[6];
    float* out = (float*)d_out;

    dim3 grid(NROWS / TILE);   // 1024 blocks
    dim3 block(256);           // 8 waves of 32
    fuzzy_fused_kernel<<<grid, block, 0, stream>>>(x, means, sigmas, W1, b1, W2, b2, out);
}